// BootstrappedCrossEntropy_42056319762859
// MI455X (gfx1250) — compile-verified
//
#include <hip/hip_runtime.h>
#include <math.h>

// Problem constants (match reference): B=8, C=19, H=512, W=1024
#define Bsz 8
#define Csz 19
#define HW   524288            // 512*1024 = 2^19
#define BHW  4194304           // 8*HW
#define CHW  ((size_t)Csz * HW)

#define NBINS_BIG 4096
#define HIST_BLOCKS 512
#define SUM_BLOCKS 512

typedef float  v2f __attribute__((ext_vector_type(2)));
typedef float  v4f __attribute__((ext_vector_type(4)));

// Workspace layout (bytes):
//   [0, 16MB)   : float loss[BHW]
//   [+16KB)     : unsigned hist[4096]
//   [+256B]     : Ctrl
//   [+2KB]      : float psum[512]
//   [+2KB]      : unsigned pcnt[512]

struct Ctrl {
  unsigned k;            // current rank from the top (1-based)
  unsigned prefix_mask;  // known-high-bits mask of the k-th largest value
  unsigned prefix_val;   // known-high-bits value
  unsigned tk_bits;      // final exact bit pattern of TK
  unsigned num;          // debug
};

// ---------------------------------------------------------------------------
// init: zero histogram, compute num from step (double, Python eval order)
// ---------------------------------------------------------------------------
__global__ __launch_bounds__(256) void bce_init(const int* __restrict__ step_ptr,
                                                unsigned* __restrict__ hist,
                                                Ctrl* __restrict__ ctrl) {
  for (int i = threadIdx.x; i < NBINS_BIG; i += 256) hist[i] = 0u;
  if (threadIdx.x == 0) {
    int step = step_ptr[0];
    double mom = pow(0.99998, (double)step);
    double mx  = (mom > 0.15) ? mom : 0.15;
    double dnum = (((0.15 * 8.0) * 512.0) * 1024.0) * mx;   // K*b*h*w*max(...)
    long long n = (long long)dnum;                          // int() truncation
    if (n < 1) n = 1;
    if (n > BHW) n = BHW;
    ctrl->k = (unsigned)n;
    ctrl->prefix_mask = 0u;
    ctrl->prefix_val  = 0u;
    ctrl->tk_bits     = 0u;
    ctrl->num         = (unsigned)n;
  }
}

// ---------------------------------------------------------------------------
// loss: two pixels per thread, NT B64 channel loads (stream-once 318MB, keep
// L2 for the loss array). Target-logit captured via cndmask (no gather).
// ---------------------------------------------------------------------------
__global__ __launch_bounds__(256) void bce_loss(const float* __restrict__ pred,
                                                const int* __restrict__ tgt,
                                                float* __restrict__ loss) {
  int q  = blockIdx.x * 256 + threadIdx.x;      // pair index; grid = BHW/2
  int p  = q << 1;                              // even pixel index
  int b  = p >> 19;                             // p / HW
  int hw = p & (HW - 1);                        // even -> 8B aligned
  const float* base = pred + (size_t)b * CHW + hw;

  __builtin_prefetch(base, 0, 0);               // gfx1250: global_prefetch_b8

  long long tt = __builtin_nontemporal_load((const long long*)tgt + q);
  int t0 = (int)tt;
  int t1 = (int)(tt >> 32);

  v2f v[Csz];
  float m0 = -3.4e38f, m1 = -3.4e38f;
  float x0 = 0.f, x1 = 0.f;
#pragma unroll
  for (int c = 0; c < Csz; ++c) {
    v2f x = __builtin_nontemporal_load((const v2f*)(base + (size_t)c * HW));
    v[c] = x;
    m0 = fmaxf(m0, x.x);
    m1 = fmaxf(m1, x.y);
    x0 = (c == t0) ? x.x : x0;                  // v_cndmask
    x1 = (c == t1) ? x.y : x1;
  }
  float s0 = 0.f, s1 = 0.f;
#pragma unroll
  for (int c = 0; c < Csz; ++c) {
    s0 += __expf(v[c].x - m0);
    s1 += __expf(v[c].y - m1);
  }
  v2f out;
  out.x = m0 + __logf(s0) - x0;
  out.y = m1 + __logf(s1) - x1;
  *(v2f*)(loss + p) = out;                      // RT store: stays in L2
}

// ---------------------------------------------------------------------------
// hist: LDS-privatized histogram. Data is staged into LDS with the gfx1250
// async path: double-buffered GLOBAL_LOAD_ASYNC_TO_LDS_B32 + S_WAIT_ASYNCCNT.
// Per-lane LDS slot -> no cross-thread dependency, no barrier needed in loop.
// BHW % (HIST_BLOCKS*256) == 0, so the loop trip count is wave-uniform.
// ---------------------------------------------------------------------------
__global__ __launch_bounds__(256) void bce_hist(const float* __restrict__ loss,
                                                const Ctrl* __restrict__ ctrl,
                                                unsigned* __restrict__ hist,
                                                int shift, int nbins) {
  __shared__ unsigned lh[NBINS_BIG];
  __shared__ float stage[2][256];
  for (int i = threadIdx.x; i < nbins; i += 256) lh[i] = 0u;
  __syncthreads();

  unsigned mask = ctrl->prefix_mask;            // uniform -> scalar load
  unsigned val  = ctrl->prefix_val;
  const int stride = gridDim.x * 256;
  const int tid = threadIdx.x;

  // Flat LDS pointer low 32 bits == allocation-relative LDS byte offset
  // (flat-LDS aperture: addr[31:0] is the DS address) -> async VDST operand.
  unsigned lds0 = (unsigned)(unsigned long long)(const void*)&stage[0][tid];
  unsigned lds1 = (unsigned)(unsigned long long)(const void*)&stage[1][tid];

  int i = blockIdx.x * 256 + tid;               // always < BHW (131072 <= BHW)
  {
    const float* g = loss + i;
    asm volatile("global_load_async_to_lds_b32 %0, %1, off"
                 : : "v"(lds0), "v"(g) : "memory");
  }
  int buf = 0;
  while (i < BHW) {
    int inext = i + stride;
    int iprf  = (inext < BHW) ? inext : i;      // tail: dummy re-stage
    const float* g = loss + iprf;
    unsigned ldsn = buf ? lds0 : lds1;
    asm volatile("global_load_async_to_lds_b32 %0, %1, off"
                 : : "v"(ldsn), "v"(g) : "memory");
    asm volatile("s_wait_asynccnt 0x1" ::: "memory");   // older buffer landed
    float L = stage[buf][tid];
    unsigned u = __float_as_uint(L);            // loss >= 0 -> bits monotone
    if ((u & mask) == val)
      atomicAdd(&lh[(u >> shift) & (unsigned)(nbins - 1)], 1u);
    buf ^= 1;
    i = inext;
  }
  asm volatile("s_wait_asynccnt 0x0" ::: "memory");     // drain dummy
  __syncthreads();
  for (int j = threadIdx.x; j < nbins; j += 256) {
    unsigned c = lh[j];
    if (c) atomicAdd(&hist[j], c);
  }
}

// ---------------------------------------------------------------------------
// select: single block. Suffix-scan histogram from the top bin, find the bin
// containing rank k, update (prefix, k), re-zero hist for the next pass.
// ---------------------------------------------------------------------------
__global__ __launch_bounds__(256) void bce_select(unsigned* __restrict__ hist,
                                                  Ctrl* __restrict__ ctrl,
                                                  int shift, int nbins,
                                                  int is_final) {
  __shared__ unsigned psc[256];
  __shared__ unsigned s_bin, s_krem;
  const int T = 256;
  int items = nbins / T;                        // 16 (4096 bins) or 1 (256)

  unsigned loc[16];
  unsigned lsum = 0;
  for (int j = 0; j < items; ++j) {
    int r = threadIdx.x * items + j;            // r counts from the TOP bin
    unsigned h = hist[nbins - 1 - r];
    loc[j] = h;
    lsum += h;
  }
  unsigned k = ctrl->k;
  if (threadIdx.x == 0) { s_bin = 0u; s_krem = 1u; }   // safety defaults
  psc[threadIdx.x] = lsum;
  __syncthreads();

  for (int off = 1; off < T; off <<= 1) {       // inclusive Hillis-Steele
    unsigned t = (threadIdx.x >= off) ? psc[threadIdx.x - off] : 0u;
    __syncthreads();
    psc[threadIdx.x] += t;
    __syncthreads();
  }
  unsigned cum = (threadIdx.x == 0) ? 0u : psc[threadIdx.x - 1];

  for (int j = 0; j < items; ++j) {
    unsigned nc = cum + loc[j];
    if (cum < k && k <= nc) {                   // exactly one (thread,j) hits
      s_bin  = (unsigned)(nbins - 1 - (threadIdx.x * items + j));
      s_krem = k - cum;
    }
    cum = nc;
  }
  __syncthreads();

  if (threadIdx.x == 0) {
    unsigned pv = ctrl->prefix_val | (s_bin << shift);
    ctrl->prefix_val  = pv;
    ctrl->prefix_mask |= ((unsigned)(nbins - 1)) << shift;
    ctrl->k = s_krem;
    if (is_final) ctrl->tk_bits = pv;           // exact TK bit pattern
  }
  for (int i = threadIdx.x; i < nbins; i += T) hist[i] = 0u;  // reset
}

// ---------------------------------------------------------------------------
// sum: deterministic two-stage reduction of loss where bits >= tk_bits
// (== loss >= TK including ties, matching the reference mask). B128 loads.
// ---------------------------------------------------------------------------
__global__ __launch_bounds__(256) void bce_sum(const float* __restrict__ loss,
                                               const Ctrl* __restrict__ ctrl,
                                               float* __restrict__ psum,
                                               unsigned* __restrict__ pcnt) {
  __shared__ float    ss[256];
  __shared__ unsigned sc[256];
  unsigned tk = ctrl->tk_bits;
  float s = 0.f;
  unsigned c = 0u;
  const int n4 = BHW / 4;
  int stride = gridDim.x * 256;
  for (int i = blockIdx.x * 256 + threadIdx.x; i < n4; i += stride) {
    v4f L = ((const v4f*)loss)[i];
    if (__float_as_uint(L.x) >= tk) { s += L.x; c += 1u; }
    if (__float_as_uint(L.y) >= tk) { s += L.y; c += 1u; }
    if (__float_as_uint(L.z) >= tk) { s += L.z; c += 1u; }
    if (__float_as_uint(L.w) >= tk) { s += L.w; c += 1u; }
  }
  ss[threadIdx.x] = s;
  sc[threadIdx.x] = c;
  __syncthreads();
  for (int off = 128; off > 0; off >>= 1) {
    if (threadIdx.x < off) {
      ss[threadIdx.x] += ss[threadIdx.x + off];
      sc[threadIdx.x] += sc[threadIdx.x + off];
    }
    __syncthreads();
  }
  if (threadIdx.x == 0) {
    psum[blockIdx.x] = ss[0];
    pcnt[blockIdx.x] = sc[0];
  }
}

__global__ __launch_bounds__(512) void bce_final(const float* __restrict__ psum,
                                                 const unsigned* __restrict__ pcnt,
                                                 float* __restrict__ out) {
  __shared__ float    ss[512];
  __shared__ unsigned sc[512];
  ss[threadIdx.x] = psum[threadIdx.x];
  sc[threadIdx.x] = pcnt[threadIdx.x];
  __syncthreads();
  for (int off = 256; off > 0; off >>= 1) {
    if (threadIdx.x < off) {
      ss[threadIdx.x] += ss[threadIdx.x + off];
      sc[threadIdx.x] += sc[threadIdx.x + off];
    }
    __syncthreads();
  }
  if (threadIdx.x == 0) out[0] = ss[0] / (float)sc[0];
}

// ---------------------------------------------------------------------------
extern "C" void kernel_launch(void* const* d_in, const int* in_sizes, int n_in,
                              void* d_out, int out_size, void* d_ws, size_t ws_size,
                              hipStream_t stream) {
  (void)in_sizes; (void)n_in; (void)out_size; (void)ws_size;
  const float* pred = (const float*)d_in[0];
  const int*   tgt  = (const int*)d_in[1];
  const int*   step = (const int*)d_in[2];
  float* out = (float*)d_out;

  char* ws = (char*)d_ws;
  float*    loss = (float*)ws;
  unsigned* hist = (unsigned*)(ws + (size_t)BHW * 4);
  Ctrl*     ctrl = (Ctrl*)(ws + (size_t)BHW * 4 + NBINS_BIG * 4);
  float*    psum = (float*)(ws + (size_t)BHW * 4 + NBINS_BIG * 4 + 256);
  unsigned* pcnt = (unsigned*)((char*)psum + SUM_BLOCKS * 4);

  bce_init  <<<1, 256, 0, stream>>>(step, hist, ctrl);
  bce_loss  <<<BHW / 2 / 256, 256, 0, stream>>>(pred, tgt, loss);

  bce_hist  <<<HIST_BLOCKS, 256, 0, stream>>>(loss, ctrl, hist, 20, 4096);
  bce_select<<<1, 256, 0, stream>>>(hist, ctrl, 20, 4096, 0);
  bce_hist  <<<HIST_BLOCKS, 256, 0, stream>>>(loss, ctrl, hist, 8, 4096);
  bce_select<<<1, 256, 0, stream>>>(hist, ctrl, 8, 4096, 0);
  bce_hist  <<<HIST_BLOCKS, 256, 0, stream>>>(loss, ctrl, hist, 0, 256);
  bce_select<<<1, 256, 0, stream>>>(hist, ctrl, 0, 256, 1);

  bce_sum   <<<SUM_BLOCKS, 256, 0, stream>>>(loss, ctrl, psum, pcnt);
  bce_final <<<1, 512, 0, stream>>>(psum, pcnt, out);
}